// TransLoss_79757542686995
// MI455X (gfx1250) — compile-verified
//
#include <hip/hip_runtime.h>
#include <math.h>

// Problem constants (match reference)
#define B_   8
#define T_   128
#define U_   64
#define U1_  65
#define V_   1024
#define ROWS_ (B_ * T_ * U1_)   // 66560 softmax rows

// ---------------------------------------------------------------------------
// Kernel 1: per-row logsumexp over V=1024, emitting lp_blank / lp_label.
// One wave per row, grid-strided; each wave double-buffers its row through
// LDS using CDNA5 async global->LDS loads (ASYNCcnt-tracked), so HBM data
// lands in LDS without a VGPR round-trip while the previous row is reduced.
// ---------------------------------------------------------------------------

__device__ __forceinline__ void issue_row_async(const float* __restrict__ g,
                                                float* lds_buf, int lane)
{
    // LDS byte offset = low 32 bits of the generic pointer to __shared__.
    unsigned lds0 = (unsigned)(unsigned long long)(lds_buf) + (unsigned)lane * 16u;
    const float* g0 = g + lane * 4;
#pragma unroll
    for (int c = 0; c < 8; ++c) {
        // 32 lanes x 16B = 512B per op; 8 ops move the full 4KB row.
        asm volatile("global_load_async_to_lds_b128 %0, %1, off"
                     :: "v"(lds0 + (unsigned)c * 512u), "v"(g0 + c * 128)
                     : "memory");
    }
}

__global__ __launch_bounds__(256, 1) void rnnt_rowlse(
    const float* __restrict__ pred,     // [B,T,U+1,V]
    const int*   __restrict__ target,   // [B,U]
    float*       __restrict__ lpb,      // [B,T,U+1]
    float*       __restrict__ lpl)      // [B,T,U]
{
    __shared__ float lbuf[8 * 2 * V_];  // 8 waves x double-buffered 4KB rows = 64KB

    const int lane = threadIdx.x & 31;
    const int wave = threadIdx.x >> 5;
    const int gw   = blockIdx.x * 8 + wave;   // global wave id
    const int nw   = gridDim.x * 8;           // wave stride

    float* buf0 = &lbuf[wave * 2 * V_];
    float* buf1 = buf0 + V_;
    float* bufs[2] = { buf0, buf1 };
    int cur = 0;

    int r = gw;
    if (r < ROWS_) {
        issue_row_async(pred + (size_t)r * V_, bufs[0], lane);
    }

    for (; r < ROWS_; r += nw) {
        const int rn = r + nw;
        if (rn < ROWS_) {
            // Distance-2 prefetch: pull HBM lines for row r+2*nw into GL2
            // ahead of the async engine (one 128B line per lane).
            const int r2 = rn + nw;
            if (r2 < ROWS_)
                __builtin_prefetch(pred + (size_t)r2 * V_ + lane * 32, 0, 0);
            issue_row_async(pred + (size_t)rn * V_, bufs[cur ^ 1], lane);
            // Async loads complete in order: <=8 outstanding means the 8 ops
            // of the *current* buffer have landed in LDS.
            asm volatile("s_wait_asynccnt 8" ::: "memory");
        } else {
            asm volatile("s_wait_asynccnt 0" ::: "memory");
        }

        const float* buf = bufs[cur];

        // 32 elems/lane from LDS (bank-conflict-free b128 reads).
        float4 v[8];
#pragma unroll
        for (int c = 0; c < 8; ++c)
            v[c] = *(const float4*)(buf + c * 128 + lane * 4);

        float m = v[0].x;
#pragma unroll
        for (int c = 0; c < 8; ++c) {
            m = fmaxf(m, v[c].x); m = fmaxf(m, v[c].y);
            m = fmaxf(m, v[c].z); m = fmaxf(m, v[c].w);
        }
#pragma unroll
        for (int off = 16; off > 0; off >>= 1)
            m = fmaxf(m, __shfl_xor(m, off, 32));

        float s = 0.f;
#pragma unroll
        for (int c = 0; c < 8; ++c) {
            s += __expf(v[c].x - m); s += __expf(v[c].y - m);
            s += __expf(v[c].z - m); s += __expf(v[c].w - m);
        }
#pragma unroll
        for (int off = 16; off > 0; off >>= 1)
            s += __shfl_xor(s, off, 32);

        const float lse = m + __logf(s);

        if (lane == 0) {
            const int b   = r / (T_ * U1_);
            const int rem = r % (T_ * U1_);
            const int t   = rem / U1_;
            const int u   = rem % U1_;
            lpb[r] = buf[0] - lse;                     // blank = vocab index 0
            if (u < U_) {
                const int tgt = target[b * U_ + u];    // 1..V-1
                lpl[(b * T_ + t) * U_ + u] = buf[tgt] - lse;
            }
        }
        cur ^= 1;
    }
}

// ---------------------------------------------------------------------------
// Kernel 2: RNNT alpha DP per batch, anti-diagonal wavefront, alpha in LDS.
// alpha[t][u] = logaddexp(alpha[t-1][u] + lpb[t-1][u],
//                         alpha[t][u-1] + lpl[t][u-1])
// ---------------------------------------------------------------------------
__global__ __launch_bounds__(96, 1) void rnnt_dp(
    const float* __restrict__ lpb,        // [B,T,U+1]
    const float* __restrict__ lpl,        // [B,T,U]
    const int*   __restrict__ pred_len,   // [B]
    const int*   __restrict__ target_len, // [B]
    float*       __restrict__ loss)       // [B]
{
    __shared__ float alpha[T_ * U1_];     // 128*65*4 = 33.3 KB
    const int b   = blockIdx.x;
    const int tid = threadIdx.x;
    const float* LPB = lpb + (size_t)b * T_ * U1_;
    const float* LPL = lpl + (size_t)b * T_ * U_;

    if (tid == 0) {                        // t = 0 row: cumsum of labels
        float acc = 0.f;
        alpha[0] = 0.f;
        for (int u = 1; u <= U_; ++u) {
            acc += LPL[u - 1];
            alpha[u] = acc;
        }
    }
    __syncthreads();

    for (int d = 1; d <= (T_ - 1) + U_; ++d) {   // 191 anti-diagonals
        const int u = tid;
        if (u <= U_) {
            const int t = d - u;
            if (t >= 1 && t <= T_ - 1) {
                const float top = alpha[(t - 1) * U1_ + u] + LPB[(t - 1) * U1_ + u];
                float a;
                if (u == 0) {
                    a = top;
                } else {
                    const float left = alpha[t * U1_ + (u - 1)] + LPL[t * U_ + (u - 1)];
                    const float mx = fmaxf(top, left);
                    const float mn = fminf(top, left);
                    a = mx + log1pf(__expf(mn - mx));
                }
                alpha[t * U1_ + u] = a;
            }
        }
        __syncthreads();
    }

    if (tid == 0) {
        const int tl = pred_len[b];
        const int ul = target_len[b];
        const float ll = alpha[(tl - 1) * U1_ + ul] + LPB[(tl - 1) * U1_ + ul];
        loss[b] = -ll;
    }
}

// ---------------------------------------------------------------------------
// Kernel 3: mean over the B per-batch losses.
// ---------------------------------------------------------------------------
__global__ void rnnt_mean(const float* __restrict__ loss, float* __restrict__ out)
{
    if (threadIdx.x == 0 && blockIdx.x == 0) {
        float s = 0.f;
        for (int i = 0; i < B_; ++i) s += loss[i];
        out[0] = s * (1.0f / B_);
    }
}

// ---------------------------------------------------------------------------
extern "C" void kernel_launch(void* const* d_in, const int* in_sizes, int n_in,
                              void* d_out, int out_size, void* d_ws, size_t ws_size,
                              hipStream_t stream)
{
    (void)in_sizes; (void)n_in; (void)out_size; (void)ws_size;
    const float* pred       = (const float*)d_in[0];  // [B,T,U+1,V] f32
    const int*   target     = (const int*)  d_in[1];  // [B,U] i32
    const int*   pred_len   = (const int*)  d_in[2];  // [B] i32
    const int*   target_len = (const int*)  d_in[3];  // [B] i32

    float* ws   = (float*)d_ws;
    float* lpb  = ws;                       // ROWS_ floats
    float* lpl  = lpb + ROWS_;              // B*T*U floats
    float* loss = lpl + (size_t)B_ * T_ * U_; // B floats

    // 1024 blocks x 8 waves = 8192 waves; ~8 rows/wave (double buffer amortizes)
    rnnt_rowlse<<<1024, 256, 0, stream>>>(pred, target, lpb, lpl);
    rnnt_dp<<<B_, 96, 0, stream>>>(lpb, lpl, pred_len, target_len, loss);
    rnnt_mean<<<1, 1, 0, stream>>>(loss, (float*)d_out);
}